// MessagePassingBonded_25512105738358
// MI455X (gfx1250) — compile-verified
//
#include <hip/hip_runtime.h>
#include <math.h>
#include <stdint.h>

#define N_NODES 200000
#define N_EDGES 800000
#define F_IN    64
#define HID     128
#define NLAYERS 3

typedef __attribute__((ext_vector_type(2))) float v2f;
typedef __attribute__((ext_vector_type(8))) float v8f;

// ---------------------------------------------------------------------------
// Tensor Data Mover availability.
//   ROCm 7.2 / clang-22 : 5-arg builtin (g0 v4u, g1 v8i, v4i, v4i, cpol)
//   amdgpu-toolchain / clang-23+: 6-arg builtin (g0, g1, v4i, v4i, v8i, cpol)
// ---------------------------------------------------------------------------
#if defined(__gfx1250__) && __has_builtin(__builtin_amdgcn_tensor_load_to_lds) && \
    __has_builtin(__builtin_amdgcn_s_wait_tensorcnt)
#define TDM_OK 1
#else
#define TDM_OK 0
#endif

#if TDM_OK
typedef unsigned int tdm_v4u __attribute__((ext_vector_type(4)));
typedef int          tdm_v8i __attribute__((ext_vector_type(8)));
typedef int          tdm_v4i __attribute__((ext_vector_type(4)));

// Load a 16-row x width_dw-col f32 tile (row-major, row stride = width_dw)
// into LDS; TDM pad fields insert 4 DWORDs every 2^(pad_code+1) DWORDs so the
// LDS row stride becomes width_dw + 4 (bank-conflict-free for the WMMA reads).
__device__ __forceinline__ void tdm_load_tile16(const float* gsrc, float* lds_dst,
                                                unsigned width_dw, unsigned pad_code) {
    unsigned long long ga = (unsigned long long)(uintptr_t)gsrc;
    unsigned la = (unsigned)(uintptr_t)lds_dst;   // low 32 bits = LDS byte offset
    tdm_v4u g0;
    g0[0] = 1u;                                   // count=1 (valid descriptor)
    g0[1] = la;                                   // lds_addr
    g0[2] = (unsigned)(ga & 0xFFFFFFFFu);         // global_addr[31:0]
    g0[3] = (unsigned)((ga >> 32) & 0x01FFFFFFu)  // global_addr[56:32]
          | (2u << 30);                           // type=2 ("image")
    tdm_v8i g1;
    g1[0] = (int)((2u << 16)                      // data_size = 4 bytes
          | (1u << 20)                            // pad_enable
          | (pad_code << 22)                      // pad_interval
          | (3u << 25));                          // pad_amount = 4 DWORDs
    g1[1] = (int)(width_dw << 16);                // tensor_dim0[15:0] @ bits 63:48
    g1[2] = (int)(((width_dw >> 16) & 0xFFFFu)    // tensor_dim0[31:16]
          | (((unsigned)N_NODES & 0xFFFFu) << 16)); // tensor_dim1[15:0]
    g1[3] = (int)((((unsigned)N_NODES >> 16) & 0xFFFFu) // tensor_dim1[31:16]
          | (width_dw << 16));                    // tile_dim0
    g1[4] = 16;                                   // tile_dim1 = 16 rows
    g1[5] = (int)width_dw;                        // tensor_dim0_stride[31:0]
    g1[6] = 0;
    g1[7] = 0;
    tdm_v4i z4 = {0, 0, 0, 0};
#if __clang_major__ >= 23
    tdm_v8i z8 = {0, 0, 0, 0, 0, 0, 0, 0};
    __builtin_amdgcn_tensor_load_to_lds(g0, g1, z4, z4, z8, 0);
#else
    __builtin_amdgcn_tensor_load_to_lds(g0, g1, z4, z4, 0);
#endif
}
#endif

// ---------------------------------------------------------------------------
// utility: zero a float buffer (element count multiple of 4)
// ---------------------------------------------------------------------------
__global__ void zero_f32(float* __restrict__ p, size_t n4) {
    size_t i = (size_t)blockIdx.x * blockDim.x + threadIdx.x;
    size_t stride = (size_t)gridDim.x * blockDim.x;
    float4* p4 = (float4*)p;
    for (; i < n4; i += stride) p4[i] = make_float4(0.f, 0.f, 0.f, 0.f);
}

// ---------------------------------------------------------------------------
// in-degree (float counts via atomics), then in-place reciprocal
// ---------------------------------------------------------------------------
__global__ void degree_kernel(const int* __restrict__ dst, float* __restrict__ deg) {
    int e = blockIdx.x * blockDim.x + threadIdx.x;
    if (e < N_EDGES) atomicAdd(&deg[dst[e]], 1.0f);
}

__global__ void invdeg_kernel(float* __restrict__ deg) {
    int i = blockIdx.x * blockDim.x + threadIdx.x;
    if (i < N_NODES) {
        float d = deg[i];
        deg[i] = (d > 0.0f) ? (1.0f / d) : 0.0f;
    }
}

// ---------------------------------------------------------------------------
// edge scatter: h_sum[dst] += h[src]   (one wave32 per edge, float4 per lane)
// ---------------------------------------------------------------------------
__global__ __launch_bounds__(256) void scatter_add(
    const float* __restrict__ h, const int* __restrict__ src,
    const int* __restrict__ dst, float* __restrict__ hsum) {
    int e = blockIdx.x * 8 + (threadIdx.x >> 5);
    if (e >= N_EDGES) return;
    int lane = threadIdx.x & 31;
    int s = src[e];
    int d = dst[e];
    float4 v = *(const float4*)(h + (size_t)s * HID + lane * 4);
    float* o = hsum + (size_t)d * HID + lane * 4;
    atomicAdd(o + 0, v.x);
    atomicAdd(o + 1, v.y);
    atomicAdd(o + 2, v.z);
    atomicAdd(o + 3, v.w);
}

// ---------------------------------------------------------------------------
// input layer: h = tanh(x @ w_in + b_in)
// grid.x = N_NODES/16 M-tiles, 8 waves per block = 8 N-tiles of 16 (HID=128)
// V_WMMA_F32_16X16X4_F32, K=64 -> 16 steps, two interleaved C-chains
// ---------------------------------------------------------------------------
__global__ __launch_bounds__(256) void gemm_in_tanh(
    const float* __restrict__ x, const float* __restrict__ w,
    const float* __restrict__ b, float* __restrict__ h) {
    __shared__ float As[16 * 68];                 // row stride 68: conflict-free

    const int tid  = threadIdx.x;
    const int lane = tid & 31;
    const int wave = tid >> 5;
    const int row0 = blockIdx.x * 16;
    const int n0   = wave * 16;
    const int m    = lane & 15;
    const int hi   = lane >> 4;                   // 0: K pair {0,1}, 1: {2,3}

#if TDM_OK
    if (wave == 0) {                              // one wave drives the DMA
        tdm_load_tile16(x + (size_t)row0 * F_IN, As, F_IN, /*pad 64 DW*/ 5u);
        __builtin_amdgcn_s_wait_tensorcnt(0);
    }
#else
    {   // cooperative A-tile load: 16 x 64 floats, float4 per thread
        int r = tid >> 4;                         // 0..15
        int c = (tid & 15) * 4;                   // 0..60
        float4 v = *(const float4*)(x + (size_t)(row0 + r) * F_IN + c);
        *(float4*)(&As[r * 68 + c]) = v;
    }
#endif

    // preload B fragments (weights are tiny & L2-hot)
    v2f bf[16];
#pragma unroll
    for (int s = 0; s < 16; ++s) {
        int k   = s * 4 + hi * 2;
        int col = n0 + m;
        bf[s].x = w[(k + 0) * HID + col];
        bf[s].y = w[(k + 1) * HID + col];
    }
    __syncthreads();

    v8f acc0 = {}, acc1 = {};                     // two chains for WMMA ILP
#pragma unroll
    for (int s = 0; s < 16; s += 2) {
        int k0 = s * 4 + hi * 2;
        int k1 = (s + 1) * 4 + hi * 2;
        v2f a0 = *(const v2f*)(&As[m * 68 + k0]);
        v2f a1 = *(const v2f*)(&As[m * 68 + k1]);
        acc0 = __builtin_amdgcn_wmma_f32_16x16x4_f32(
            false, a0, false, bf[s], (short)0, acc0, false, false);
        acc1 = __builtin_amdgcn_wmma_f32_16x16x4_f32(
            false, a1, false, bf[s + 1], (short)0, acc1, false, false);
    }

    float bias = b[n0 + m];
#pragma unroll
    for (int v = 0; v < 8; ++v) {
        int row = row0 + v + 8 * hi;              // C/D layout: M = v + 8*(lane/16)
        h[(size_t)row * HID + n0 + m] = tanhf(acc0[v] + acc1[v] + bias);
    }
}

// ---------------------------------------------------------------------------
// SAGE layer: out = relu(h @ w_self + diag(inv_deg) * (h_sum @ w_neigh) + b)
// (row scaling commuted into the epilogue -> A tiles are pure row copies,
//  staged by the Tensor Data Mover with hardware LDS padding)
// K=128 -> 32 WMMA steps per operand, two independent accumulators
// ---------------------------------------------------------------------------
__global__ __launch_bounds__(256) void sage_gemm(
    const float* __restrict__ h, const float* __restrict__ hsum,
    const float* __restrict__ invdeg, const float* __restrict__ wsf,
    const float* __restrict__ wnb, const float* __restrict__ b,
    float* __restrict__ out) {
    __shared__ float As[16 * 132];                // row stride 132: conflict-free
    __shared__ float An[16 * 132];
    __shared__ float sInv[16];

    const int tid  = threadIdx.x;
    const int lane = tid & 31;
    const int wave = tid >> 5;
    const int row0 = blockIdx.x * 16;
    const int n0   = wave * 16;
    const int m    = lane & 15;
    const int hi   = lane >> 4;

    if (tid < 16) sInv[tid] = invdeg[row0 + tid];

#if TDM_OK
    if (wave == 0) {                              // one wave drives both DMAs
        tdm_load_tile16(h    + (size_t)row0 * HID, As, HID, /*pad 128 DW*/ 6u);
        tdm_load_tile16(hsum + (size_t)row0 * HID, An, HID, /*pad 128 DW*/ 6u);
        __builtin_amdgcn_s_wait_tensorcnt(0);     // both loads, in order
    }
#else
    {   // cooperative A-tile loads: 16 x 128 each, 8 floats/thread/matrix
        int r = tid >> 4;                         // 0..15
        int c = (tid & 15) * 8;                   // 0..120
        const float* hr = h + (size_t)(row0 + r) * HID + c;
        *(float4*)(&As[r * 132 + c])     = *(const float4*)(hr);
        *(float4*)(&As[r * 132 + c + 4]) = *(const float4*)(hr + 4);
        const float* sr = hsum + (size_t)(row0 + r) * HID + c;
        *(float4*)(&An[r * 132 + c])     = *(const float4*)(sr);
        *(float4*)(&An[r * 132 + c + 4]) = *(const float4*)(sr + 4);
    }
#endif

    // preload both weight-fragment sets into registers (64 v2f = 128 VGPRs)
    v2f bs[32], bn[32];
#pragma unroll
    for (int s = 0; s < 32; ++s) {
        int k   = s * 4 + hi * 2;
        int col = n0 + m;
        bs[s].x = wsf[(k + 0) * HID + col];
        bs[s].y = wsf[(k + 1) * HID + col];
        bn[s].x = wnb[(k + 0) * HID + col];
        bn[s].y = wnb[(k + 1) * HID + col];
    }
    __syncthreads();

    v8f accs = {}, accn = {};
#pragma unroll
    for (int s = 0; s < 32; ++s) {
        int k = s * 4 + hi * 2;
        v2f a0 = *(const v2f*)(&As[m * 132 + k]);
        v2f a1 = *(const v2f*)(&An[m * 132 + k]);
        accs = __builtin_amdgcn_wmma_f32_16x16x4_f32(
            false, a0, false, bs[s], (short)0, accs, false, false);
        accn = __builtin_amdgcn_wmma_f32_16x16x4_f32(
            false, a1, false, bn[s], (short)0, accn, false, false);
    }

    float bias = b[n0 + m];
#pragma unroll
    for (int v = 0; v < 8; ++v) {
        int rl  = v + 8 * hi;                     // local row 0..15
        float val = accs[v] + accn[v] * sInv[rl] + bias;
        out[(size_t)(row0 + rl) * HID + n0 + m] = val > 0.0f ? val : 0.0f;
    }
}

// ---------------------------------------------------------------------------
// launch
// ---------------------------------------------------------------------------
extern "C" void kernel_launch(void* const* d_in, const int* in_sizes, int n_in,
                              void* d_out, int out_size, void* d_ws, size_t ws_size,
                              hipStream_t stream) {
    const float* x       = (const float*)d_in[0];
    const int*   src     = (const int*)d_in[1];
    const int*   dst     = (const int*)d_in[2];
    const float* w_in    = (const float*)d_in[3];
    const float* b_in    = (const float*)d_in[4];
    const float* w_self  = (const float*)d_in[5];
    const float* w_neigh = (const float*)d_in[6];
    const float* b_sage  = (const float*)d_in[7];
    float* out = (float*)d_out;

    float* ws_f  = (float*)d_ws;
    float* h_a   = ws_f;                                   // N_NODES*HID
    float* h_sum = ws_f + (size_t)N_NODES * HID;           // N_NODES*HID
    float* ideg  = ws_f + 2 * (size_t)N_NODES * HID;       // N_NODES

    const size_t feat_elems = (size_t)N_NODES * HID;

    // degree -> inverse degree (in place in ideg)
    zero_f32<<<512, 256, 0, stream>>>(ideg, (size_t)N_NODES / 4);
    degree_kernel<<<(N_EDGES + 255) / 256, 256, 0, stream>>>(dst, ideg);
    invdeg_kernel<<<(N_NODES + 255) / 256, 256, 0, stream>>>(ideg);

    // input projection + tanh
    gemm_in_tanh<<<N_NODES / 16, 256, 0, stream>>>(x, w_in, b_in, h_a);

    const float* cur = h_a;
    for (int i = 0; i < NLAYERS; ++i) {
        float* nxt = (i % 2 == 0) ? out : h_a;   // i=0 -> out, i=1 -> h_a, i=2 -> out
        zero_f32<<<4096, 256, 0, stream>>>(h_sum, feat_elems / 4);
        scatter_add<<<N_EDGES / 8, 256, 0, stream>>>(cur, src, dst, h_sum);
        sage_gemm<<<N_NODES / 16, 256, 0, stream>>>(
            cur, h_sum, ideg,
            w_self  + (size_t)i * HID * HID,
            w_neigh + (size_t)i * HID * HID,
            b_sage  + (size_t)i * HID,
            nxt);
        cur = nxt;
    }
}